// CombinedLayer_15625091023062
// MI455X (gfx1250) — compile-verified
//
#include <hip/hip_runtime.h>

// ---------------------------------------------------------------------------
// GNN layer for MI455X (gfx1250, wave32).
// Phase A: edge scatter (global_atomic_add_f32 + ordered-uint atomic max).
//          Whole working set (~58 MB) is L2-resident (192 MB) -> L2/atomic bound.
// Phase B: per-node 2-way attention blend + residual (wave-per-node).
// Phase C: fused 3-layer MLP, V_WMMA_F32_16X16X4_F32 (full f32 precision),
//          tiles staged into LDS by the Tensor Data Mover (TENSOR_LOAD_TO_LDS)
//          using TDM hardware padding to produce the stride-65 LDS layout.
// ---------------------------------------------------------------------------

typedef __attribute__((ext_vector_type(2))) float v2f;
typedef __attribute__((ext_vector_type(8))) float v8f;
typedef __attribute__((ext_vector_type(4))) unsigned int u32x4;
typedef __attribute__((ext_vector_type(8))) int i32x8;
typedef __attribute__((ext_vector_type(4))) int i32x4;

#define LDW 65  // 64 + 1 DWORD pad per row (TDM pad_interval=64dw, amount=1dw)

// Monotonic float <-> uint mapping so unsigned atomicMax == float max
// (handles negatives; sentinel 0u is below every mapped real value).
__device__ __forceinline__ unsigned f2o(float f) {
  unsigned b = __float_as_uint(f);
  return (b & 0x80000000u) ? ~b : (b | 0x80000000u);
}
__device__ __forceinline__ float o2f(unsigned o) {
  return (o & 0x80000000u) ? __uint_as_float(o & 0x7fffffffu)
                           : __uint_as_float(~o);
}

// ------------------------- Phase A: edge scatter ---------------------------
// One wave per directed edge. edge_index layout: [2][E] int32 (lo, hi).
__global__ void scatter_kernel(const float* __restrict__ h,
                               const int* __restrict__ ei,
                               float* __restrict__ agg_sum,
                               unsigned* __restrict__ agg_max,
                               unsigned* __restrict__ deg, int E) {
  int wave = (blockIdx.x * blockDim.x + threadIdx.x) >> 5;
  int lane = threadIdx.x & 31;
  if (wave >= 2 * E) return;
  // First E waves: (r,c)=(lo[e],hi[e]); second E: (r,c)=(hi[e],lo[e]).
  // ei[wave] is lo[e] in the first half and hi[e] in the second half.
  int r = ei[wave];
  int c = ei[wave < E ? wave + E : wave - E];

  float v0 = h[c * 64 + lane];
  float v1 = h[c * 64 + lane + 32];

  unsafeAtomicAdd(&agg_sum[r * 64 + lane], v0);       // global_atomic_add_f32
  unsafeAtomicAdd(&agg_sum[r * 64 + lane + 32], v1);
  atomicMax(&agg_max[r * 64 + lane], f2o(v0));        // global_atomic_max_u32
  atomicMax(&agg_max[r * 64 + lane + 32], f2o(v1));
  if (lane == 0) atomicAdd(&deg[r], 1u);
}

// --------------------- Phase B: attention blend ----------------------------
__global__ void attn_kernel(const float* __restrict__ h,
                            const float* __restrict__ agg_sum,
                            const unsigned* __restrict__ agg_max,
                            const unsigned* __restrict__ deg,
                            const float* __restrict__ att_w,
                            const float* __restrict__ att_b,
                            float* __restrict__ x, int n) {
  int node = (blockIdx.x * blockDim.x + threadIdx.x) >> 5;
  int lane = threadIdx.x & 31;
  if (node >= n) return;
  int base = node * 64;

  float s0 = agg_sum[base + lane];
  float s1 = agg_sum[base + lane + 32];
  float m0 = 0.f, m1 = 0.f;
  if (deg[node] > 0u) {
    m0 = o2f(agg_max[base + lane]);
    m1 = o2f(agg_max[base + lane + 32]);
  }
  // att_w is [128][2] row-major; concat order: [sum(64), max(64)].
  float sc0 = s0 * att_w[lane * 2] + s1 * att_w[(lane + 32) * 2] +
              m0 * att_w[(64 + lane) * 2] + m1 * att_w[(96 + lane) * 2];
  float sc1 = s0 * att_w[lane * 2 + 1] + s1 * att_w[(lane + 32) * 2 + 1] +
              m0 * att_w[(64 + lane) * 2 + 1] + m1 * att_w[(96 + lane) * 2 + 1];
#pragma unroll
  for (int off = 16; off; off >>= 1) {
    sc0 += __shfl_xor(sc0, off, 32);
    sc1 += __shfl_xor(sc1, off, 32);
  }
  sc0 += att_b[0];
  sc1 += att_b[1];
  float mx = fmaxf(sc0, sc1);
  float e0 = __expf(sc0 - mx), e1 = __expf(sc1 - mx);
  float inv = 1.0f / (e0 + e1);
  float w0 = e0 * inv, w1v = e1 * inv;

  x[base + lane] = h[base + lane] + w0 * s0 + w1v * m0;
  x[base + lane + 32] = h[base + lane + 32] + w0 * s1 + w1v * m1;
}

// ------------------- TDM: 2-D f32 tile Global -> LDS -----------------------
// Loads a (rows<=64) x 64 f32 tile with row stride 64 from global into LDS,
// padding 1 DWORD after every 64 DWORDs (=> LDS row stride 65). Rows beyond
// `rows_avail` read as zero (ISA 8.2 OOB rule) -> free tail-block zero-fill.
// D# per CDNA5 ISA 8.3-8.5; groups 2/3 and the trailing group are zero
// (2-D tile, dims >=2 unused). 6-arg builtin form (amdgpu-toolchain lane).
__device__ __forceinline__ void tdm_load_tile64(const float* gsrc,
                                                unsigned lds_byte_off,
                                                unsigned rows_avail) {
  unsigned long long ga = (unsigned long long)(uintptr_t)gsrc;
  u32x4 g0;
  g0[0] = 1u;                                   // count=1, user descriptor
  g0[1] = lds_byte_off;                         // lds_addr
  g0[2] = (unsigned)(ga & 0xffffffffu);         // global_addr[31:0]
  g0[3] = (unsigned)((ga >> 32) & 0x01ffffffu)  // global_addr[56:32]
          | (2u << 30);                         // type=2 ("image")
  i32x8 g1;
  g1[0] = (int)((2u << 16)      // data_size = 4 bytes
                | (1u << 20)    // pad_enable
                | (5u << 22));  // pad_interval: 64 DWORDs (pad_amount=0 -> 1dw)
  g1[1] = (int)(64u << 16);     // tensor_dim0 = 64 (bits[79:48], low half)
  g1[2] = (int)((rows_avail & 0xffffu) << 16);  // dim0 hi=0 | tensor_dim1 lo
  g1[3] = (int)((rows_avail >> 16) | (64u << 16));  // dim1 hi | tile_dim0=64
  g1[4] = 64;                   // tile_dim1 = 64, tile_dim2 = 0
  g1[5] = 64;                   // tensor_dim0_stride = 64 (row stride)
  g1[6] = 0;                    // dim0_stride hi | dim1_stride lo
  g1[7] = 0;
  i32x4 z4 = {0, 0, 0, 0};
  i32x8 z8 = {0, 0, 0, 0, 0, 0, 0, 0};
  __builtin_amdgcn_tensor_load_to_lds(g0, g1, z4, z4, z8, 0);
}

// --------------------- Phase C: fused 3-layer MLP --------------------------
// Fragment layouts per CDNA5 ISA 7.12.2:
//  A 16x4 : lanes 0-15 -> (M=lane, K=0/1), lanes 16-31 -> (M=lane-16, K=2/3)
//  B 4x16 : VGPR0 lanes0-15=K0, lanes16-31=K2; VGPR1 lanes0-15=K1, lanes16-31=K3
//  C 16x16: VGPR j: lanes0-15 -> M=j, N=lane; lanes16-31 -> M=j+8, N=lane-16
__device__ __forceinline__ void wmma_tiles(const float* src, const float* w,
                                           int lane, int wid, v8f* acc) {
  const int half = lane >> 4;
  const int l15 = lane & 15;
  const int kofs = half * 2;
#pragma unroll
  for (int ti = 0; ti < 2; ++ti) {
    const int t = wid * 2 + ti;
    const int tm = (t >> 2) << 4;
    const int tn = (t & 3) << 4;
    const int arow = tm + l15;
    const int bcol = tn + l15;
    v8f c = {};
#pragma unroll
    for (int kb = 0; kb < 64; kb += 4) {
      v2f a, b;
      a[0] = src[arow * LDW + kb + kofs];
      a[1] = src[arow * LDW + kb + kofs + 1];
      b[0] = w[(kb + kofs) * LDW + bcol];
      b[1] = w[(kb + kofs + 1) * LDW + bcol];
      c = __builtin_amdgcn_wmma_f32_16x16x4_f32(false, a, false, b, (short)0, c,
                                                false, false);
    }
    acc[ti] = c;
  }
}

__device__ __forceinline__ void store_tiles_relu(const v8f* acc, float* dst,
                                                 const float* bias, int lane,
                                                 int wid) {
  const int half = lane >> 4;
  const int l15 = lane & 15;
#pragma unroll
  for (int ti = 0; ti < 2; ++ti) {
    const int t = wid * 2 + ti;
    const int tm = (t >> 2) << 4;
    const int tn = (t & 3) << 4;
    const float bv = bias[tn + l15];
#pragma unroll
    for (int j = 0; j < 8; ++j)
      dst[(tm + j + half * 8) * LDW + tn + l15] = fmaxf(acc[ti][j] + bv, 0.f);
  }
}

__global__ __launch_bounds__(256) void mlp_kernel(
    const float* x, const float* __restrict__ w1, const float* __restrict__ b1,
    const float* __restrict__ w2, const float* __restrict__ b2,
    const float* __restrict__ w3, const float* __restrict__ b3, float* out,
    int n) {
  __shared__ float xs[64 * LDW];
  __shared__ float ys[64 * LDW];
  __shared__ float wsm[64 * LDW];
  __shared__ float bsm[64];
  const int tid = threadIdx.x;
  const int lane = tid & 31, wid = tid >> 5;
  const int r0 = blockIdx.x * 64;

  const unsigned xs_off = (unsigned)(uintptr_t)&xs[0];
  const unsigned ws_off = (unsigned)(uintptr_t)&wsm[0];

  // Stage x tile + layer-1 weights via the Tensor Data Mover (wave 0 issues;
  // TDM ignores EXEC and issues once per wave, so guard at wave granularity).
  if (wid == 0) {
    tdm_load_tile64(x + (size_t)r0 * 64, xs_off, (unsigned)(n - r0));
    tdm_load_tile64(w1, ws_off, 64u);
    __builtin_amdgcn_s_wait_tensorcnt(0);
  }
  if (tid < 64) bsm[tid] = b1[tid];
  __syncthreads();

  v8f acc[2];
  wmma_tiles(xs, wsm, lane, wid, acc);          // layer 1
  store_tiles_relu(acc, ys, bsm, lane, wid);
  __syncthreads();

  if (wid == 0) {
    tdm_load_tile64(w2, ws_off, 64u);
    __builtin_amdgcn_s_wait_tensorcnt(0);
  }
  if (tid < 64) bsm[tid] = b2[tid];
  __syncthreads();

  wmma_tiles(ys, wsm, lane, wid, acc);          // layer 2
  store_tiles_relu(acc, xs, bsm, lane, wid);
  __syncthreads();

  if (wid == 0) {
    tdm_load_tile64(w3, ws_off, 64u);
    __builtin_amdgcn_s_wait_tensorcnt(0);
  }
  if (tid < 64) bsm[tid] = b3[tid];
  __syncthreads();

  wmma_tiles(xs, wsm, lane, wid, acc);          // layer 3 (no relu) -> global
  const int half = lane >> 4, l15 = lane & 15;
#pragma unroll
  for (int ti = 0; ti < 2; ++ti) {
    const int t = wid * 2 + ti;
    const int tm = (t >> 2) << 4;
    const int tn = (t & 3) << 4;
    const float bv = bsm[tn + l15];
#pragma unroll
    for (int j = 0; j < 8; ++j) {
      int gr = r0 + tm + j + half * 8;
      if (gr < n) out[gr * 64 + tn + l15] = acc[ti][j] + bv;
    }
  }
}

// ---------------------------------------------------------------------------
extern "C" void kernel_launch(void* const* d_in, const int* in_sizes, int n_in,
                              void* d_out, int out_size, void* d_ws,
                              size_t ws_size, hipStream_t stream) {
  const float* h = (const float*)d_in[0];
  const int* ei = (const int*)d_in[1];
  const float* att_w = (const float*)d_in[2];
  const float* att_b = (const float*)d_in[3];
  const float* w1 = (const float*)d_in[4];
  const float* b1 = (const float*)d_in[5];
  const float* w2 = (const float*)d_in[6];
  const float* b2 = (const float*)d_in[7];
  const float* w3 = (const float*)d_in[8];
  const float* b3 = (const float*)d_in[9];
  float* out = (float*)d_out;
  const int n = in_sizes[0] / 64;   // 50000 nodes
  const int E = in_sizes[1] / 2;    // 800000 undirected edges

  // Workspace: [agg_sum f32 n*64][agg_max u32 n*64][deg u32 n] (~25.8 MB).
  float* agg_sum = (float*)d_ws;
  unsigned* agg_max = (unsigned*)((char*)d_ws + (size_t)n * 64 * 4);
  unsigned* deg = (unsigned*)((char*)d_ws + (size_t)n * 64 * 8);
  size_t init_bytes = (size_t)n * 64 * 8 + (size_t)n * 4;
  (void)hipMemsetAsync(d_ws, 0, init_bytes, stream);  // 0u == ordered(-inf)

  // Phase A: 1.6M directed edges, 1 wave each, 8 waves per block.
  int dirEdges = 2 * E;
  scatter_kernel<<<(dirEdges + 7) / 8, 256, 0, stream>>>(h, ei, agg_sum,
                                                         agg_max, deg, E);
  // Phase B: 1 wave per node; writes x into d_out (reused as MLP input).
  attn_kernel<<<(n + 7) / 8, 256, 0, stream>>>(h, agg_sum, agg_max, deg, att_w,
                                               att_b, out, n);
  // Phase C: 64-row tiles; each block reads its rows into LDS (via TDM)
  // before overwriting them, so in-place out->out is safe.
  mlp_kernel<<<(n + 63) / 64, 256, 0, stream>>>(out, w1, b1, w2, b2, w3, b3,
                                                out, n);
}